// MeanAggregator_506806141515
// MI455X (gfx1250) — compile-verified
//
#include <hip/hip_runtime.h>
#include <stdint.h>

#ifndef __has_builtin
#define __has_builtin(x) 0
#endif

// Problem constants (from the reference): R=4, B=2048, S=10, U=20000, D=128
#define R_ 4
#define B_ 2048
#define S_ 10
#define U_ 20000
#define D_ 128
#define ROWS (R_ * B_)          // 8192 output rows
#define WAVES_PER_BLOCK 4       // wave32: blockDim = (32, 4); LDS = 4*10*128*4B = 20 KB

typedef __attribute__((ext_vector_type(4))) unsigned int v4u;
typedef __attribute__((ext_vector_type(4))) int          v4i;
typedef __attribute__((ext_vector_type(8))) int          v8i;
typedef __attribute__((ext_vector_type(4))) float        f4;

// LDS-address-space float4: dereference lowers to ds_load_b128
typedef __attribute__((address_space(3))) const f4 lds_cf4;

#define RFL(x) __builtin_amdgcn_readfirstlane((int)(x))

__global__ __launch_bounds__(32 * WAVES_PER_BLOCK)
void mean_agg_tdm_gather(const int* __restrict__ nidx,
                         const float* __restrict__ embed,
                         float* __restrict__ out)
{
    __shared__ float lds_buf[WAVES_PER_BLOCK * S_ * D_];

    const int lane = threadIdx.x;                       // 0..31
    const int wv   = threadIdx.y;                       // 0..WAVES_PER_BLOCK-1
    const int row  = blockIdx.x * WAVES_PER_BLOCK + wv; // one wave per output row

    // ---- 1) load the 10 neighbor ids for this row (wave-uniform values) ----
    int idx[S_];
#pragma unroll
    for (int s = 0; s < S_; ++s) idx[s] = nidx[row * S_ + s];

    // Flat shared pointer truncated to 32 bits == LDS byte offset (ISA §10.2)
    unsigned lds_off =
        (unsigned)RFL((unsigned)(uintptr_t)(void*)&lds_buf[wv * (S_ * D_)]);

#if __has_builtin(__builtin_amdgcn_tensor_load_to_lds)
    // ---- 2) build Tensor DMA descriptor (gather mode, 16-bit row indices)
    //         and issue the DMA immediately so it overlaps the dedup math ----
    {
        const uint64_t ga = (uint64_t)(uintptr_t)embed;

        v4u g0;
        g0[0] = (unsigned)RFL(0x80000001u);          // count=1 | gather_mode=1 | 16b idx
        g0[1] = lds_off;                             // lds_addr (bytes)
        g0[2] = (unsigned)RFL((uint32_t)ga);         // global_addr[31:0]
        g0[3] = (unsigned)RFL(((uint32_t)(ga >> 32) & 0x01FFFFFFu) | (2u << 30)); // addr[56:32] | type=2

        v8i g1;
        g1[0] = RFL(2 << 16);                // workgroup_mask=0 | data_size=4B (code 2)
        g1[1] = RFL(D_ << 16);               // atomic_barrier_addr=0 | tensor_dim0[15:0]=128
        g1[2] = RFL((U_ & 0xFFFF) << 16);    // tensor_dim0[31:16]=0 | tensor_dim1[15:0]=20000
        g1[3] = RFL(D_ << 16);               // tensor_dim1[31:16]=0 | tile_dim0=128
        g1[4] = RFL(S_);                     // tile_dim1 = 10 valid gather indices | tile_dim2=0
        g1[5] = RFL(D_);                     // tensor_dim0_stride = 128 (row pitch, elems)
        g1[6] = 0;
        g1[7] = 0;

        v4i g2, g3;                          // sixteen 16-bit row indices (10 valid)
        g2[0] = RFL((idx[0] & 0xFFFF) | (idx[1] << 16));
        g2[1] = RFL((idx[2] & 0xFFFF) | (idx[3] << 16));
        g2[2] = RFL((idx[4] & 0xFFFF) | (idx[5] << 16));
        g2[3] = RFL((idx[6] & 0xFFFF) | (idx[7] << 16));
        g3[0] = RFL((idx[8] & 0xFFFF) | (idx[9] << 16));
        g3[1] = 0; g3[2] = 0; g3[3] = 0;

#if defined(__clang_major__) && (__clang_major__ >= 23)
        v8i gz = {0, 0, 0, 0, 0, 0, 0, 0};
        __builtin_amdgcn_tensor_load_to_lds(g0, g1, g2, g3, gz, 0);
#else
        __builtin_amdgcn_tensor_load_to_lds(g0, g1, g2, g3, 0);
#endif
    }
#endif

    // ---- 3) dedup while the TDM gather is in flight:
    //         keep first occurrence only; n = |unique set| ----
    bool keep[S_];
    int n = 0;
#pragma unroll
    for (int s = 0; s < S_; ++s) {
        bool k = true;
#pragma unroll
        for (int t = 0; t < S_; ++t)
            if (t < s && idx[t] == idx[s]) k = false;
        keep[s] = k;
        n += k ? 1 : 0;
    }
    const float inv = 1.0f / (float)n;

    f4 acc = {0.f, 0.f, 0.f, 0.f};

#if __has_builtin(__builtin_amdgcn_tensor_load_to_lds)
    // ---- 4) wait for the DMA, then reduce the 10x128 tile from LDS ----
    __builtin_amdgcn_s_wait_tensorcnt(0);

    // Launder the LDS offset so the compiler can't prove the buffer is
    // unwritten (the TDM write is invisible to it), then rebuild an
    // address_space(3) pointer -> ds_load_b128.
    asm volatile("" : "+v"(lds_off) : : "memory");
    lds_cf4* rows4 = (lds_cf4*)(uintptr_t)lds_off;   // 10 rows x 32 float4
#pragma unroll
    for (int s = 0; s < S_; ++s) {
        if (keep[s]) {
            f4 v = rows4[s * (D_ / 4) + lane];
            acc += v;
        }
    }
#else
    // Fallback: direct global gather (no TDM builtin on this toolchain)
#pragma unroll
    for (int s = 0; s < S_; ++s) {
        if (keep[s]) {
            const f4* src = (const f4*)&embed[(size_t)idx[s] * D_];
            acc += src[lane];
        }
    }
#endif

    // ---- 5) scale by 1/n_unique and store one float4 per lane ----
    f4* o = (f4*)&out[(size_t)row * D_];
    o[lane] = acc * inv;
}

extern "C" void kernel_launch(void* const* d_in, const int* in_sizes, int n_in,
                              void* d_out, int out_size, void* d_ws, size_t ws_size,
                              hipStream_t stream)
{
    const int*   nidx  = (const int*)d_in[0];    // [R, B, S] int32
    const float* embed = (const float*)d_in[1];  // [U, D] float32
    float*       out   = (float*)d_out;          // [R*B, D] float32

    dim3 block(32, WAVES_PER_BLOCK);
    dim3 grid(ROWS / WAVES_PER_BLOCK);           // 8192 / 4 = 2048 blocks
    hipLaunchKernelGGL(mean_agg_tdm_gather, grid, block, 0, stream,
                       nidx, embed, out);
}